// SAN_39805756900157
// MI455X (gfx1250) — compile-verified
//
#include <hip/hip_runtime.h>
#include <hip/hip_bf16.h>
#include <math.h>

#define N_NODES 8192
#define E_EDGES 131072
#define DIM     768
#define HEADS   12
#define DKH     64
#define DVH     64
#define NBD     64
#define RELD    32
#define ARCD    32

typedef float v2f __attribute__((ext_vector_type(2)));
typedef float v8f __attribute__((ext_vector_type(8)));

#define BLK_K   16          // K panel depth staged in LDS
#define KP_STR  160         // words per kpair row: 64 cols * 2 + 32 pad
                            // (160 % 64 == 32 -> half-wave groups hit disjoint banks)

// ---------------------------------------------------------------------------
// FP32 WMMA GEMM: C[M,Nc] = A[M,K] @ B[K,Nc]   (row-major)
// blockDim = (32, 8): 8 waves; each wave computes a 32x64 tile of C
// (2 M-tiles x 4 N-tiles = 8 WMMA accumulators, B fragments reused 2x).
// Block covers 256 rows x 64 cols. Requires M%256==0, Nc%64==0, K%16==0.
//
// B panels (16x64) are cooperatively staged into LDS (double-buffered) in a
// K-paired layout Bs[kpair][col][2] so each B fragment is ONE ds_load_b64
// directly into a WMMA operand pair (no repack movs, conflict-free banks).
//
// V_WMMA_F32_16X16X4_F32 operand layout (ISA 7.12.2):
//   A (16x4, 2 VGPRs): lane m=lane&15, VGPR p holds K = p + 2*(lane>>4)
//   B (4x16, 2 VGPRs): lane n=lane&15, VGPR p holds K = p + 2*(lane>>4)
//   C/D (16x16, 8 VGPRs): VGPR r holds M = r + 8*(lane>>4), N = lane&15
// ---------------------------------------------------------------------------
__global__ void gemm_f32_wmma(const float* __restrict__ A,
                              const float* __restrict__ B,
                              float* __restrict__ C,
                              int M, int Nc, int K) {
  const int lane = threadIdx.x;
  const int wave = threadIdx.y;
  const int row0 = blockIdx.y * 256 + wave * 32;
  const int col0 = blockIdx.x * 64;
  const int mloc = lane & 15;
  const int half = lane >> 4;

  // staging decomposition: 256 threads move a 16x64 panel, 4 floats each.
  // thread -> kpair (2 consecutive K rows) and a 2-column chunk.
  const int tid = wave * 32 + lane;
  const int skp = tid >> 5;          // 0..7  (K pair within panel)
  const int sc0 = (tid & 31) << 1;   // 0..62 (column, 2-col chunk)

  __shared__ float Bs[2][8 * KP_STR];

  v8f a0c0 = {}, a0c1 = {}, a0c2 = {}, a0c3 = {};
  v8f a1c0 = {}, a1c1 = {}, a1c2 = {}, a1c3 = {};

  const float* Arow0 = A + (size_t)(row0 + mloc) * K;
  const float* Arow1 = Arow0 + (size_t)16 * K;

  auto stage = [&](int k0, int buf) {
    const float* g = B + (size_t)(k0 + 2 * skp) * Nc + col0 + sc0;
    float2 r0 = *(const float2*)g;        // row k0+2*skp,   cols sc0..sc0+1
    float2 r1 = *(const float2*)(g + Nc); // row k0+2*skp+1, cols sc0..sc0+1
    float4 w;                             // paired layout: (c,p0),(c,p1),(c+1,p0),(c+1,p1)
    w.x = r0.x; w.y = r1.x; w.z = r0.y; w.w = r1.y;
    *(float4*)&Bs[buf][skp * KP_STR + sc0 * 2] = w;
  };

  stage(0, 0);
  __syncthreads();

  const int nPanels = K / BLK_K;
  for (int p = 0; p < nPanels; ++p) {
    const int k0 = p * BLK_K;
    if (p + 1 < nPanels) stage(k0 + BLK_K, (p + 1) & 1);
    // pull the A streams ahead (global_prefetch_b8; speculative, OOB-safe)
    __builtin_prefetch(Arow0 + k0 + 4 * BLK_K, 0, 1);
    __builtin_prefetch(Arow1 + k0 + 4 * BLK_K, 0, 1);

    const float* bs = &Bs[p & 1][0];
    #pragma unroll
    for (int kk = 0; kk < BLK_K; kk += 4) {
      v2f a0, a1;
      a0.x = Arow0[k0 + kk + 2 * half + 0];
      a0.y = Arow0[k0 + kk + 2 * half + 1];
      a1.x = Arow1[k0 + kk + 2 * half + 0];
      a1.y = Arow1[k0 + kk + 2 * half + 1];
      const float* bp = bs + ((kk >> 1) + half) * KP_STR + mloc * 2;
      v2f b0 = *(const v2f*)(bp +  0);   // single ds_load_b64 per fragment
      v2f b1 = *(const v2f*)(bp + 32);
      v2f b2 = *(const v2f*)(bp + 64);
      v2f b3 = *(const v2f*)(bp + 96);
      a0c0 = __builtin_amdgcn_wmma_f32_16x16x4_f32(false, a0, false, b0, (short)0, a0c0, false, false);
      a0c1 = __builtin_amdgcn_wmma_f32_16x16x4_f32(false, a0, false, b1, (short)0, a0c1, false, false);
      a0c2 = __builtin_amdgcn_wmma_f32_16x16x4_f32(false, a0, false, b2, (short)0, a0c2, false, false);
      a0c3 = __builtin_amdgcn_wmma_f32_16x16x4_f32(false, a0, false, b3, (short)0, a0c3, false, false);
      a1c0 = __builtin_amdgcn_wmma_f32_16x16x4_f32(false, a1, false, b0, (short)0, a1c0, false, false);
      a1c1 = __builtin_amdgcn_wmma_f32_16x16x4_f32(false, a1, false, b1, (short)0, a1c1, false, false);
      a1c2 = __builtin_amdgcn_wmma_f32_16x16x4_f32(false, a1, false, b2, (short)0, a1c2, false, false);
      a1c3 = __builtin_amdgcn_wmma_f32_16x16x4_f32(false, a1, false, b3, (short)0, a1c3, false, false);
    }
    __syncthreads();
  }

  float* Cc = C + (size_t)col0 + mloc;
  #pragma unroll
  for (int r = 0; r < 8; ++r) {
    size_t m0 = (size_t)(row0 + r + 8 * half);
    size_t m1 = m0 + 16;
    Cc[m0 * Nc +  0] = a0c0[r];
    Cc[m0 * Nc + 16] = a0c1[r];
    Cc[m0 * Nc + 32] = a0c2[r];
    Cc[m0 * Nc + 48] = a0c3[r];
    Cc[m1 * Nc +  0] = a1c0[r];
    Cc[m1 * Nc + 16] = a1c1[r];
    Cc[m1 * Nc + 32] = a1c2[r];
    Cc[m1 * Nc + 48] = a1c3[r];
  }
}

// ----------------------- monotonic float <-> uint for atomicMax ------------
__device__ __forceinline__ unsigned mono_enc(float f) {
  unsigned u = __float_as_uint(f);
  return (u & 0x80000000u) ? ~u : (u | 0x80000000u);
}
__device__ __forceinline__ float mono_dec(unsigned e) {
  unsigned u = (e & 0x80000000u) ? (e & 0x7FFFFFFFu) : ~e;
  return __uint_as_float(u);
}

// ------------------- edge scores: one wave per (edge, head) -----------------
__global__ void edge_scores(const float* __restrict__ q,
                            const float* __restrict__ kk,
                            const float* __restrict__ relk_tab,
                            const float* __restrict__ arck_tab,
                            const int* __restrict__ rel_idx,
                            const int* __restrict__ arc_idx,
                            const int* __restrict__ src,
                            const int* __restrict__ tgt,
                            float* __restrict__ scores,
                            unsigned* __restrict__ smax_bits) {
  const int lane = threadIdx.x;
  const long wid = (long)blockIdx.x * 8 + threadIdx.y;
  if (wid >= (long)E_EDGES * HEADS) return;
  const int e = (int)(wid / HEADS);
  const int h = (int)(wid % HEADS);
  const int s = src[e], t = tgt[e];
  const float rk0 = relk_tab[rel_idx[e] * RELD + lane];   // d = lane      (<32)
  const float rk1 = arck_tab[arc_idx[e] * ARCD + lane];   // d = lane + 32
  const float* qp = q  + (size_t)t * (HEADS * DKH) + h * DKH;
  const float* kp = kk + (size_t)s * (HEADS * DKH) + h * DKH;
  float acc = qp[lane]      * (kp[lane]      + rk0)
            + qp[lane + 32] * (kp[lane + 32] + rk1);
  #pragma unroll
  for (int off = 16; off > 0; off >>= 1)
    acc += __shfl_xor(acc, off, 32);
  if (lane == 0) {
    float sc = acc * 0.125f;                      // 1/sqrt(DK=64)
    scores[(size_t)e * HEADS + h] = sc;
    atomicMax(&smax_bits[(size_t)t * HEADS + h], mono_enc(sc));
  }
}

// ----------------- exp(score - max) and denominator accumulation -----------
__global__ void edge_ex(const int* __restrict__ tgt,
                        float* __restrict__ scores,   // in: score, out: exp
                        const unsigned* __restrict__ smax_bits,
                        float* __restrict__ denom) {
  long i = (long)blockIdx.x * blockDim.x + threadIdx.x;
  if (i >= (long)E_EDGES * HEADS) return;
  int e = (int)(i / HEADS), h = (int)(i % HEADS);
  int t = tgt[e];
  float m = mono_dec(smax_bits[(size_t)t * HEADS + h]);
  float ex = __expf(scores[i] - m);
  scores[i] = ex;
  atomicAdd(&denom[(size_t)t * HEADS + h], ex);
}

// --------- weighted value aggregation: one block per edge, 256 threads over 768 values -------
__global__ void edge_agg(const int* __restrict__ src,
                         const int* __restrict__ tgt,
                         const float* __restrict__ ex,
                         const float* __restrict__ denom,
                         const float* __restrict__ v,
                         const float* __restrict__ relv_tab,
                         const float* __restrict__ arcv_tab,
                         const int* __restrict__ rel_idx,
                         const int* __restrict__ arc_idx,
                         float* __restrict__ agg) {
  const int e = blockIdx.x;
  const int s = src[e], t = tgt[e];
  const int ri = rel_idx[e], ai = arc_idx[e];
  for (int idx = threadIdx.x; idx < HEADS * DVH; idx += blockDim.x) {
    int h = idx >> 6, dv = idx & 63;
    float attn = ex[(size_t)e * HEADS + h] /
                 (denom[(size_t)t * HEADS + h] + 1e-16f);
    float rv = (dv < RELD) ? relv_tab[ri * RELD + dv]
                           : arcv_tab[ai * ARCD + dv - RELD];
    float val = v[(size_t)s * (HEADS * DVH) + idx] + rv;
    atomicAdd(&agg[(size_t)t * (HEADS * DVH) + idx], attn * val);
  }
}

// --------------------------- neighbor aggregation --------------------------
__global__ void edge_nb(const int* __restrict__ src,
                        const int* __restrict__ tgt,
                        const float* __restrict__ nb,
                        float* __restrict__ nbagg) {
  long i = (long)blockIdx.x * blockDim.x + threadIdx.x;
  if (i >= (long)E_EDGES * NBD) return;
  int e = (int)(i >> 6), j = (int)(i & 63);
  atomicAdd(&nbagg[(size_t)tgt[e] * NBD + j], nb[(size_t)src[e] * NBD + j]);
}

// ------------------------------- gate MLP ----------------------------------
__global__ void gate_kernel(const float* __restrict__ inp,
                            const float* __restrict__ nbagg,
                            const float* __restrict__ Wg,
                            float* __restrict__ gate) {
  int i = blockIdx.x * blockDim.x + threadIdx.x;
  if (i >= N_NODES * HEADS) return;
  int n = i / HEADS, h = i % HEADS;
  float acc = 0.f;
  const float* x = inp + (size_t)n * DIM;
  for (int d = 0; d < DIM; ++d) acc += x[d] * Wg[d * HEADS + h];
  const float* y = nbagg + (size_t)n * NBD;
  for (int j = 0; j < NBD; ++j) acc += y[j] * Wg[(DIM + j) * HEADS + h];
  gate[i] = 1.f / (1.f + __expf(-acc));
}

// --------------------------- apply gate in place ---------------------------
__global__ void scale_kernel(float* __restrict__ agg,
                             const float* __restrict__ gate) {
  long i = (long)blockIdx.x * blockDim.x + threadIdx.x;
  if (i >= (long)N_NODES * HEADS * DVH) return;
  int n = (int)(i / (HEADS * DVH));
  int h = ((int)(i % (HEADS * DVH))) >> 6;
  agg[i] *= gate[n * HEADS + h];
}

extern "C" void kernel_launch(void* const* d_in, const int* in_sizes, int n_in,
                              void* d_out, int out_size, void* d_ws, size_t ws_size,
                              hipStream_t stream) {
  (void)in_sizes; (void)n_in; (void)out_size; (void)ws_size;
  const float* inp    = (const float*)d_in[0];
  const float* Wq     = (const float*)d_in[1];
  const float* Wk     = (const float*)d_in[2];
  const float* Wv     = (const float*)d_in[3];
  const float* Wo     = (const float*)d_in[4];
  const float* Wn     = (const float*)d_in[5];
  const float* Wg     = (const float*)d_in[6];
  const float* relk_t = (const float*)d_in[7];
  const float* arck_t = (const float*)d_in[8];
  const float* relv_t = (const float*)d_in[9];
  const float* arcv_t = (const float*)d_in[10];
  const int*   rel_i  = (const int*)d_in[11];
  const int*   arc_i  = (const int*)d_in[12];
  const int*   eidx   = (const int*)d_in[13];
  const int*   src    = eidx;            // edge_index[0]
  const int*   tgt    = eidx + E_EDGES;  // edge_index[1]
  float* out = (float*)d_out;

  char* ws = (char*)d_ws;
  size_t off = 0;
  auto alloc = [&](size_t bytes) -> void* {
    void* p = ws + off;
    off += (bytes + 255) & ~(size_t)255;
    return p;
  };
  float*    q         = (float*)alloc((size_t)N_NODES * DIM * 4);
  float*    kbuf      = (float*)alloc((size_t)N_NODES * DIM * 4);
  float*    vbuf      = (float*)alloc((size_t)N_NODES * DIM * 4);
  float*    nbbuf     = (float*)alloc((size_t)N_NODES * NBD * 4);
  float*    scores    = (float*)alloc((size_t)E_EDGES * HEADS * 4);
  unsigned* smax_bits = (unsigned*)alloc((size_t)N_NODES * HEADS * 4);
  float*    denom     = (float*)alloc((size_t)N_NODES * HEADS * 4);
  float*    agg       = (float*)alloc((size_t)N_NODES * DIM * 4);
  float*    nbagg     = (float*)alloc((size_t)N_NODES * NBD * 4);
  float*    gate      = (float*)alloc((size_t)N_NODES * HEADS * 4);

  dim3 gblk(32, 8);
  // Q, K, V, Wn projections
  gemm_f32_wmma<<<dim3(DIM / 64, N_NODES / 256), gblk, 0, stream>>>(inp, Wq, q,    N_NODES, DIM, DIM);
  gemm_f32_wmma<<<dim3(DIM / 64, N_NODES / 256), gblk, 0, stream>>>(inp, Wk, kbuf, N_NODES, DIM, DIM);
  gemm_f32_wmma<<<dim3(DIM / 64, N_NODES / 256), gblk, 0, stream>>>(inp, Wv, vbuf, N_NODES, DIM, DIM);
  gemm_f32_wmma<<<dim3(NBD / 64, N_NODES / 256), gblk, 0, stream>>>(inp, Wn, nbbuf, N_NODES, NBD, DIM);

  hipMemsetAsync(smax_bits, 0, (size_t)N_NODES * HEADS * 4, stream);
  hipMemsetAsync(denom,     0, (size_t)N_NODES * HEADS * 4, stream);
  hipMemsetAsync(agg,       0, (size_t)N_NODES * DIM * 4,   stream);
  hipMemsetAsync(nbagg,     0, (size_t)N_NODES * NBD * 4,   stream);

  long nwarp = (long)E_EDGES * HEADS;
  edge_scores<<<dim3((unsigned)((nwarp + 7) / 8)), dim3(32, 8), 0, stream>>>(
      q, kbuf, relk_t, arck_t, rel_i, arc_i, src, tgt, scores, smax_bits);
  edge_ex<<<dim3((unsigned)((nwarp + 255) / 256)), 256, 0, stream>>>(
      tgt, scores, smax_bits, denom);
  edge_agg<<<dim3(E_EDGES), 256, 0, stream>>>(
      src, tgt, scores, denom, vbuf, relv_t, arcv_t, rel_i, arc_i, agg);
  edge_nb<<<dim3((unsigned)(((long)E_EDGES * NBD + 255) / 256)), 256, 0, stream>>>(
      src, tgt, nbbuf, nbagg);
  gate_kernel<<<dim3((N_NODES * HEADS + 255) / 256), 256, 0, stream>>>(
      inp, nbagg, Wg, gate);
  scale_kernel<<<dim3((unsigned)(((long)N_NODES * DIM + 255) / 256)), 256, 0, stream>>>(
      agg, gate);
  // final output projection
  gemm_f32_wmma<<<dim3(DIM / 64, N_NODES / 256), gblk, 0, stream>>>(agg, Wo, out, N_NODES, DIM, DIM);
}